// LinkPredictor_58574763983245
// MI455X (gfx1250) — compile-verified
//
#include <hip/hip_runtime.h>

// MI455X (gfx1250) — wave32, WMMA. fp32 path via V_WMMA_F32_16X16X4_F32 to
// match the fp32 reference (RNE FMA array, full precision).

typedef __attribute__((ext_vector_type(2))) float v2f;
typedef __attribute__((ext_vector_type(8))) float v8f;

static __device__ __forceinline__ v8f wmma_f32(v2f a, v2f b, v8f c) {
    return __builtin_amdgcn_wmma_f32_16x16x4_f32(false, a, false, b, (short)0, c,
                                                 false, false);
}

// TRANSB=true : C[M,N] = A[M,K](lda) @ B[N,K](ldb)^T   (linear-layer form;
//               A and B fragments are both contiguous b64 loads)
// TRANSB=false: C[M,N] = A[M,K](lda) @ B[K,N](ldb)     (B frags = scalar pairs)
// One wave computes a 32(M) x 64(N) tile: 2 A-frags x 4 B-frags = 8 WMMAs per
// k-step of 4. Software-pipelined: next step's fragments prefetched into
// fresh registers before the current step's WMMAs.
// Requires M%32==0, N%64==0, K%4==0 (all shapes here comply).
template <bool TRANSB>
__global__ __launch_bounds__(256) void wmma_gemm(
    const float* __restrict__ A, const float* __restrict__ B,
    float* __restrict__ C, int M, int N, int K, int lda, int ldb, int ldc)
{
    const int lane = threadIdx.x & 31;
    const int wid  = (blockIdx.x * blockDim.x + threadIdx.x) >> 5;
    const int stripsN = N >> 6;                 // 64-wide strips along N
    const int tileM = wid / stripsN;
    const int tileN = wid - tileM * stripsN;
    if (tileM * 32 >= M) return;                // wave-uniform: EXEC all-1s for WMMA

    const int half = lane >> 4;                 // 0: K={0,1}, 1: K={2,3} (ISA layouts)
    const int l15  = lane & 15;

    // A fragment pointers: rows tileM*32 + l15 and +16; element kk = k + 2*half.
    const float* Ap0 = A + (size_t)(tileM * 32 + l15) * lda + 2 * half;
    const float* Ap1 = Ap0 + (size_t)16 * lda;
    // B fragment base.
    const float* Bp;
    if (TRANSB) Bp = B + (size_t)(tileN * 64 + l15) * ldb + 2 * half;  // [N,K]
    else        Bp = B + (size_t)(2 * half) * ldb + (tileN * 64 + l15); // [K,N]
    const size_t bAdv  = TRANSB ? 4 : (size_t)ldb * 4;   // per-k-step advance
    const size_t bTile = TRANSB ? (size_t)16 * ldb : 16; // between 16-wide N tiles

    v8f acc[2][4];
#pragma unroll
    for (int i = 0; i < 2; ++i)
#pragma unroll
        for (int t = 0; t < 4; ++t) acc[i][t] = v8f{0,0,0,0,0,0,0,0};

    v2f a0, a1, b[4];
    auto loadFrags = [&](v2f& ra0, v2f& ra1, v2f (&rb)[4]) {
        ra0 = *(const v2f*)Ap0;
        ra1 = *(const v2f*)Ap1;
#pragma unroll
        for (int t = 0; t < 4; ++t) {
            if (TRANSB) rb[t] = *(const v2f*)(Bp + bTile * t);       // b64
            else        rb[t] = v2f{ Bp[16 * t], Bp[ldb + 16 * t] }; // 2x b32
        }
    };

    loadFrags(a0, a1, b);                       // prologue (k = 0)
    for (int k = 0; k + 4 < K; k += 4) {
        Ap0 += 4; Ap1 += 4; Bp += bAdv;
        v2f na0, na1, nb[4];
        loadFrags(na0, na1, nb);                // prefetch k+4
#pragma unroll
        for (int t = 0; t < 4; ++t) acc[0][t] = wmma_f32(a0, b[t], acc[0][t]);
#pragma unroll
        for (int t = 0; t < 4; ++t) acc[1][t] = wmma_f32(a1, b[t], acc[1][t]);
        a0 = na0; a1 = na1;
#pragma unroll
        for (int t = 0; t < 4; ++t) b[t] = nb[t];
    }
#pragma unroll
    for (int t = 0; t < 4; ++t) acc[0][t] = wmma_f32(a0, b[t], acc[0][t]);
#pragma unroll
    for (int t = 0; t < 4; ++t) acc[1][t] = wmma_f32(a1, b[t], acc[1][t]);

    // D layout: VGPR r -> (m = r + 8*half, n = l15) within each 16x16 tile.
    const int nb0 = tileN * 64 + l15;
#pragma unroll
    for (int i = 0; i < 2; ++i) {
        const int mb = tileM * 32 + 16 * i + 8 * half;
#pragma unroll
        for (int t = 0; t < 4; ++t)
#pragma unroll
            for (int r = 0; r < 8; ++r)
                C[(size_t)(mb + r) * ldc + (nb0 + 16 * t)] = acc[i][t][r];
    }
}

// Per-edge fused MLP tail: out[e] = w2 . relu(Asrc[s] + Adst[d] + b1) + b2
// One wave (32 lanes) per edge; HID=512 -> 4 float4 per lane; wave32 reduction.
__global__ __launch_bounds__(256) void edge_mlp_kernel(
    const int* __restrict__ eidx, int E,
    const float* __restrict__ Asrc, const float* __restrict__ Adst,
    const float* __restrict__ b1, const float* __restrict__ w2,
    const float* __restrict__ b2, float* __restrict__ out)
{
    const int wid  = (blockIdx.x * blockDim.x + threadIdx.x) >> 5;
    const int lane = threadIdx.x & 31;
    if (wid >= E) return;

    const int s = eidx[wid];        // wave-uniform -> scalar loads
    const int d = eidx[E + wid];
    const float4* As = (const float4*)(Asrc + (size_t)s * 512);
    const float4* Ad = (const float4*)(Adst + (size_t)d * 512);
    const float4* Bb = (const float4*)b1;
    const float4* Ww = (const float4*)w2;

    float sum = 0.f;
#pragma unroll
    for (int j = 0; j < 4; ++j) {
        const int i = lane + 32 * j;
        float4 a = As[i], b = Ad[i], c = Bb[i], w = Ww[i];
        float x0 = fmaxf(a.x + b.x + c.x, 0.f);
        float x1 = fmaxf(a.y + b.y + c.y, 0.f);
        float x2 = fmaxf(a.z + b.z + c.z, 0.f);
        float x3 = fmaxf(a.w + b.w + c.w, 0.f);
        sum = fmaf(x0, w.x, sum);
        sum = fmaf(x1, w.y, sum);
        sum = fmaf(x2, w.z, sum);
        sum = fmaf(x3, w.w, sum);
    }
#pragma unroll
    for (int off = 16; off > 0; off >>= 1)
        sum += __shfl_xor(sum, off, 32);
    if (lane == 0) out[wid] = sum + b2[0];
}

extern "C" void kernel_launch(void* const* d_in, const int* in_sizes, int n_in,
                              void* d_out, int out_size, void* d_ws, size_t ws_size,
                              hipStream_t stream)
{
    const float* drug  = (const float*)d_in[0];   // [nDrug, 256]
    const float* prot  = (const float*)d_in[1];   // [nProt, 1280]
    const int*   eidx  = (const int*)  d_in[2];   // [2, E]
    const float* w_src = (const float*)d_in[3];   // [512, 256]
    const float* w_dst = (const float*)d_in[4];   // [512, 1280]
    const float* w1    = (const float*)d_in[5];   // [512, 1024]
    const float* b1    = (const float*)d_in[6];   // [512]
    const float* w2    = (const float*)d_in[7];   // [1, 512]
    const float* b2    = (const float*)d_in[8];   // [1]
    float* out = (float*)d_out;

    const int IN_DRUG = 256, IN_PROT = 1280, HID = 512;
    const int nDrug = in_sizes[0] / IN_DRUG;      // 20000
    const int nProt = in_sizes[1] / IN_PROT;      // 8000
    const int E     = in_sizes[2] / 2;            // 131072

    // Workspace layout (floats): ~60.5 MB total
    float* ws   = (float*)d_ws;
    float* Msrc = ws;                                  // [HID, IN_DRUG] = W1a @ w_src
    float* Mdst = Msrc + (size_t)HID * IN_DRUG;        // [HID, IN_PROT] = W1b @ w_dst
    float* Asrc = Mdst + (size_t)HID * IN_PROT;        // [nDrug, HID]
    float* Adst = Asrc + (size_t)nDrug * HID;          // [nProt, HID]

    auto blocksFor = [](int strips) { return (strips + 7) / 8; };   // 8 waves/block

    // Stage 0a: Msrc = w1[:, :512] @ w_src        [512, 256]   (C = A @ B)
    {
        const int M = HID, N = IN_DRUG, K = HID;
        wmma_gemm<false><<<blocksFor((M / 32) * (N / 64)), 256, 0, stream>>>(
            w1, w_src, Msrc, M, N, K, 2 * HID, IN_DRUG, IN_DRUG);
    }
    // Stage 0b: Mdst = w1[:, 512:] @ w_dst        [512, 1280]  (C = A @ B)
    {
        const int M = HID, N = IN_PROT, K = HID;
        wmma_gemm<false><<<blocksFor((M / 32) * (N / 64)), 256, 0, stream>>>(
            w1 + HID, w_dst, Mdst, M, N, K, 2 * HID, IN_PROT, IN_PROT);
    }
    // Stage 1: Asrc = drug_emb @ Msrc^T           [nDrug, 512] (C = A @ B^T)
    {
        const int M = nDrug, N = HID, K = IN_DRUG;
        wmma_gemm<true><<<blocksFor((M / 32) * (N / 64)), 256, 0, stream>>>(
            drug, Msrc, Asrc, M, N, K, IN_DRUG, IN_DRUG, HID);
    }
    // Stage 2: Adst = protein_emb @ Mdst^T        [nProt, 512] (C = A @ B^T)
    {
        const int M = nProt, N = HID, K = IN_PROT;
        wmma_gemm<true><<<blocksFor((M / 32) * (N / 64)), 256, 0, stream>>>(
            prot, Mdst, Adst, M, N, K, IN_PROT, IN_PROT, HID);
    }
    // Stage 3: per-edge fused ReLU-MLP tail (L2-resident gathers)
    edge_mlp_kernel<<<(E + 7) / 8, 256, 0, stream>>>(eidx, E, Asrc, Adst, b1, w2, b2, out);
}